// GcnNet_StructuralFeatures_74577812128600
// MI455X (gfx1250) — compile-verified
//
#include <hip/hip_runtime.h>
#include <cfloat>

typedef float v2f __attribute__((ext_vector_type(2)));
typedef float v8f __attribute__((ext_vector_type(8)));

#define THREADS 256

// ---------------- degree / norm ----------------

__global__ __launch_bounds__(THREADS) void k_init_deg(float* deg, int n) {
    int i = blockIdx.x * THREADS + threadIdx.x;
    if (i < n) deg[i] = 1.0f;  // self loop
}

__global__ __launch_bounds__(THREADS) void k_count_deg(const int* __restrict__ dst,
                                                       float* deg, int e) {
    int i = blockIdx.x * THREADS + threadIdx.x;
    if (i < e) atomicAdd(&deg[dst[i]], 1.0f);
}

__global__ __launch_bounds__(THREADS) void k_rsqrt(float* deg, int n) {
    int i = blockIdx.x * THREADS + threadIdx.x;
    if (i < n) deg[i] = rsqrtf(deg[i]);  // deg buffer now holds dis
}

__global__ __launch_bounds__(THREADS) void k_norm(const int* __restrict__ src,
                                                  const int* __restrict__ dst,
                                                  const float* __restrict__ dis,
                                                  float* __restrict__ norm, int e) {
    int i = blockIdx.x * THREADS + threadIdx.x;
    if (i < e) norm[i] = dis[src[i]] * dis[dst[i]];
}

// ---------------- WMMA GEMM: Out[n,64] = A[n,K] @ W[K,64] ----------------

template <int K>
__global__ __launch_bounds__(THREADS)
void k_gemm_wmma(const float* __restrict__ A, const float* __restrict__ W,
                 float* __restrict__ Out, int nRows) {
    __shared__ float As[32 * K];
    const int rowBase = blockIdx.x * 32;

    // rows are contiguous in row-major A: tile is one contiguous block
    const float4* srcp = reinterpret_cast<const float4*>(A + (size_t)rowBase * K);
    float4* dstp = reinterpret_cast<float4*>(As);
    #pragma unroll
    for (int i = threadIdx.x; i < (32 * K) / 4; i += THREADS) dstp[i] = srcp[i];
    __syncthreads();

    const int lane = threadIdx.x & 31;
    const int wv   = threadIdx.x >> 5;       // 0..7
    const int rt   = wv >> 2;                // row tile 0..1
    const int ct   = wv & 3;                 // col tile 0..3
    const int kg   = (lane >> 4) << 1;       // 0 or 2 (k-pair per half wave)
    const float* arow = &As[(rt * 16 + (lane & 15)) * K];
    const int n = ct * 16 + (lane & 15);

    v8f c = {0.f, 0.f, 0.f, 0.f, 0.f, 0.f, 0.f, 0.f};
    #pragma unroll
    for (int kk = 0; kk < K; kk += 4) {
        v2f a, b;
        a.x = arow[kk + kg];
        a.y = arow[kk + kg + 1];
        b.x = W[(size_t)(kk + kg) * 64 + n];
        b.y = W[(size_t)(kk + kg + 1) * 64 + n];
        c = __builtin_amdgcn_wmma_f32_16x16x4_f32(false, a, false, b, (short)0, c,
                                                  false, false);
    }

    const int row0 = rowBase + rt * 16 + ((lane >> 4) << 3);
    const int col  = ct * 16 + (lane & 15);
    #pragma unroll
    for (int v = 0; v < 8; ++v)
        Out[(size_t)(row0 + v) * 64 + col] = c[v];
}

// ---------------- aggregation ----------------

// H[i,f] = dis[i]^2 * T[i,f]   (self-loop term, also initializes accumulator)
__global__ __launch_bounds__(THREADS) void k_init_self(const float* __restrict__ dis,
                                                       const float* __restrict__ T,
                                                       float* __restrict__ H, int total) {
    int t = blockIdx.x * THREADS + threadIdx.x;
    if (t >= total) return;
    int i = t >> 6;
    float d = dis[i];
    H[t] = d * d * T[t];
}

__global__ __launch_bounds__(THREADS) void k_edge_scatter(const int* __restrict__ src,
                                                          const int* __restrict__ dst,
                                                          const float* __restrict__ norm,
                                                          const float* __restrict__ T,
                                                          float* __restrict__ H, int total) {
    int t = blockIdx.x * THREADS + threadIdx.x;
    if (t >= total) return;
    int e = t >> 6, f = t & 63;
    atomicAdd(&H[(size_t)dst[e] * 64 + f], norm[e] * T[(size_t)src[e] * 64 + f]);
}

__global__ __launch_bounds__(THREADS) void k_bias_relu(float* __restrict__ H,
                                                       const float* __restrict__ b, int total) {
    int t = blockIdx.x * THREADS + threadIdx.x;
    if (t >= total) return;
    H[t] = fmaxf(H[t] + b[t & 63], 0.0f);
}

// ---------------- attention pooling ----------------

__device__ inline void atomicMaxFloat(float* addr, float val) {
    int* ai = reinterpret_cast<int*>(addr);
    int old = __float_as_int(*addr);
    while (__int_as_float(old) < val) {
        int assumed = old;
        old = atomicCAS(ai, assumed, __float_as_int(val));
        if (old == assumed) break;
    }
}

__global__ __launch_bounds__(THREADS) void k_graph_init(float* m, float* z, float* g, int G) {
    int t = blockIdx.x * THREADS + threadIdx.x;
    if (t < G) m[t] = -FLT_MAX;
    else if (t < 2 * G) z[t - G] = 0.0f;
    else if (t < 66 * G) g[t - 2 * G] = 0.0f;
}

__global__ __launch_bounds__(THREADS) void k_logits_max(const float* __restrict__ clos,
                                                        const float* __restrict__ Wc,
                                                        const float* __restrict__ bc,
                                                        const int* __restrict__ batch,
                                                        float* __restrict__ logits,
                                                        float* m, int n) {
    int i = blockIdx.x * THREADS + threadIdx.x;
    if (i >= n) return;
    float s = bc[0];
    #pragma unroll
    for (int k = 0; k < 8; ++k) s += clos[(size_t)i * 8 + k] * Wc[k];
    logits[i] = s;
    atomicMaxFloat(&m[batch[i]], s);
}

__global__ __launch_bounds__(THREADS) void k_exp_sum(float* __restrict__ logits,
                                                     const float* __restrict__ m,
                                                     float* z,
                                                     const int* __restrict__ batch, int n) {
    int i = blockIdx.x * THREADS + threadIdx.x;
    if (i >= n) return;
    int b = batch[i];
    float e = expf(logits[i] - m[b]);
    logits[i] = e;
    atomicAdd(&z[b], e);
}

// counts cancels: g[b] += (e_i / z[b]) * h[i]
__global__ __launch_bounds__(THREADS) void k_pool(const float* __restrict__ logits,
                                                  const float* __restrict__ z,
                                                  const int* __restrict__ batch,
                                                  const float* __restrict__ H,
                                                  float* g, int total) {
    int t = blockIdx.x * THREADS + threadIdx.x;
    if (t >= total) return;
    int i = t >> 6, f = t & 63;
    int b = batch[i];
    atomicAdd(&g[(size_t)b * 64 + f], logits[i] / z[b] * H[t]);
}

__global__ __launch_bounds__(THREADS) void k_head(const float* __restrict__ g,
                                                  const float* __restrict__ Wa1,
                                                  const float* __restrict__ ba1,
                                                  const float* __restrict__ Wa2,
                                                  const float* __restrict__ ba2,
                                                  float* __restrict__ out, int G) {
    int gr = blockIdx.x * THREADS + threadIdx.x;
    if (gr >= G) return;
    float acc = ba2[0];
    #pragma unroll
    for (int j = 0; j < 16; ++j) {
        float s = ba1[j];
        for (int f = 0; f < 64; ++f) s += g[(size_t)gr * 64 + f] * Wa1[f * 16 + j];
        acc += fmaxf(s, 0.0f) * Wa2[j];
    }
    out[gr] = acc;
}

// ---------------- launch ----------------

extern "C" void kernel_launch(void* const* d_in, const int* in_sizes, int n_in,
                              void* d_out, int out_size, void* d_ws, size_t ws_size,
                              hipStream_t stream) {
    const float* x    = (const float*)d_in[0];
    const float* clos = (const float*)d_in[1];
    const float* W1   = (const float*)d_in[2];
    const float* b1   = (const float*)d_in[3];
    const float* W2   = (const float*)d_in[4];
    const float* b2   = (const float*)d_in[5];
    const float* W3   = (const float*)d_in[6];
    const float* b3   = (const float*)d_in[7];
    const float* Wc   = (const float*)d_in[8];
    const float* bc   = (const float*)d_in[9];
    const float* Wa1  = (const float*)d_in[10];
    const float* ba1  = (const float*)d_in[11];
    const float* Wa2  = (const float*)d_in[12];
    const float* ba2  = (const float*)d_in[13];
    const int* eidx   = (const int*)d_in[14];
    const int* batch  = (const int*)d_in[15];

    const int N  = in_sizes[15];
    const int E  = in_sizes[14] / 2;
    const int K0 = in_sizes[0] / N;  // 128
    const int G  = out_size;         // 256

    const int* srcp = eidx;
    const int* dstp = eidx + E;

    char* ws = (char*)d_ws;
    size_t off = 0;
    auto take = [&](size_t bytes) {
        char* p = ws + off;
        off = (off + bytes + 255) & ~(size_t)255;
        return p;
    };
    float* dis    = (float*)take((size_t)N * 4);        // deg -> dis
    float* norm   = (float*)take((size_t)E * 4);
    float* T      = (float*)take((size_t)N * 64 * 4);   // GEMM output
    float* H      = (float*)take((size_t)N * 64 * 4);   // aggregated features
    float* logits = (float*)take((size_t)N * 4);
    float* mbuf   = (float*)take((size_t)G * 4);
    float* zbuf   = (float*)take((size_t)G * 4);
    float* gbuf   = (float*)take((size_t)G * 64 * 4);
    (void)ws_size; (void)n_in;

    const int NF = N * 64;
    const int EF = E * 64;
    dim3 blk(THREADS);
    auto cdiv = [](int a, int b) { return (a + b - 1) / b; };

    // normalization (computed once, reused by all 3 layers)
    k_init_deg<<<cdiv(N, THREADS), blk, 0, stream>>>(dis, N);
    k_count_deg<<<cdiv(E, THREADS), blk, 0, stream>>>(dstp, dis, E);
    k_rsqrt<<<cdiv(N, THREADS), blk, 0, stream>>>(dis, N);
    k_norm<<<cdiv(E, THREADS), blk, 0, stream>>>(srcp, dstp, dis, norm, E);

    // layer 1: X[N,128] @ W1
    if (K0 == 128)
        k_gemm_wmma<128><<<N / 32, blk, 0, stream>>>(x, W1, T, N);
    else
        k_gemm_wmma<64><<<N / 32, blk, 0, stream>>>(x, W1, T, N);
    k_init_self<<<cdiv(NF, THREADS), blk, 0, stream>>>(dis, T, H, NF);
    k_edge_scatter<<<cdiv(EF, THREADS), blk, 0, stream>>>(srcp, dstp, norm, T, H, EF);
    k_bias_relu<<<cdiv(NF, THREADS), blk, 0, stream>>>(H, b1, NF);

    // layer 2
    k_gemm_wmma<64><<<N / 32, blk, 0, stream>>>(H, W2, T, N);
    k_init_self<<<cdiv(NF, THREADS), blk, 0, stream>>>(dis, T, H, NF);
    k_edge_scatter<<<cdiv(EF, THREADS), blk, 0, stream>>>(srcp, dstp, norm, T, H, EF);
    k_bias_relu<<<cdiv(NF, THREADS), blk, 0, stream>>>(H, b2, NF);

    // layer 3
    k_gemm_wmma<64><<<N / 32, blk, 0, stream>>>(H, W3, T, N);
    k_init_self<<<cdiv(NF, THREADS), blk, 0, stream>>>(dis, T, H, NF);
    k_edge_scatter<<<cdiv(EF, THREADS), blk, 0, stream>>>(srcp, dstp, norm, T, H, EF);
    k_bias_relu<<<cdiv(NF, THREADS), blk, 0, stream>>>(H, b3, NF);

    // attention pooling + head
    k_graph_init<<<cdiv(66 * G, THREADS), blk, 0, stream>>>(mbuf, zbuf, gbuf, G);
    k_logits_max<<<cdiv(N, THREADS), blk, 0, stream>>>(clos, Wc, bc, batch, logits, mbuf, N);
    k_exp_sum<<<cdiv(N, THREADS), blk, 0, stream>>>(logits, mbuf, zbuf, batch, N);
    k_pool<<<cdiv(NF, THREADS), blk, 0, stream>>>(logits, zbuf, batch, H, gbuf, NF);
    k_head<<<cdiv(G, THREADS), blk, 0, stream>>>(gbuf, Wa1, ba1, Wa2, ba2, (float*)d_out, G);
}